// Net_61684320305429
// MI455X (gfx1250) — compile-verified
//
#include <hip/hip_runtime.h>

typedef __attribute__((ext_vector_type(2))) float v2f;
typedef __attribute__((ext_vector_type(8))) float v8f;

#define N_NODES 100000
#define N_EDGES 3200000
#define IN_F    64
#define OUT_F   16
#define NODE_TILES (N_NODES / 16)   // 6250, exact

// ---------------------------------------------------------------------------
// Kernel 1: h = x @ W  via V_WMMA_F32_16X16X4_F32.
// One wave32 computes one 16(nodes) x 16(out features) tile; K=64 as 16 WMMAs.
// ---------------------------------------------------------------------------
__global__ void gcn_gemm_wmma(const float* __restrict__ x,
                              const float* __restrict__ W,
                              float* __restrict__ h) {
    const int lane = threadIdx.x & 31;
    const int wave = threadIdx.x >> 5;
    const int tile = blockIdx.x * (blockDim.x >> 5) + wave;
    if (tile >= NODE_TILES) return;   // wave-uniform guard: EXEC all-1s inside

    const int n  = lane & 15;          // M index for A rows, N index for B/C/D
    const int kb = (lane >> 4) << 1;   // K sub-offset: lanes 0-15 -> {0,1}, 16-31 -> {2,3}

    // Preload all 16 B fragments. W is [64,16] row-major (K-major), tiny & L2-hot.
    v2f bfrag[16];
#pragma unroll
    for (int k = 0; k < 16; ++k) {
        const float* wp = W + (4 * k + kb) * OUT_F + n;
        bfrag[k].x = wp[0];        // (K = 4k+kb,   N = n)
        bfrag[k].y = wp[OUT_F];    // (K = 4k+kb+1, N = n)
    }

    // A fragment source: row M = tile*16 + n of x, contiguous K pair -> v2f load.
    const float* xrow = x + (size_t)(tile * 16 + n) * IN_F;

    v8f c = {};
#pragma unroll
    for (int k = 0; k < 16; ++k) {
        v2f a = *(const v2f*)(xrow + 4 * k + kb);   // 8B-aligned
        // (neg_a, A, neg_b, B, c_mod, C, reuse_a, reuse_b)
        c = __builtin_amdgcn_wmma_f32_16x16x4_f32(false, a, false, bfrag[k],
                                                  (short)0, c, false, false);
    }

    // D layout: VGPR i -> row (i + 8*(lane>=16)), col = lane&15.
    const int rbase = tile * 16 + (lane >> 4) * 8;
#pragma unroll
    for (int i = 0; i < 8; ++i)
        h[(size_t)(rbase + i) * OUT_F + n] = c[i];
}

// ---------------------------------------------------------------------------
// Kernel 0: zero the aggregation buffer (d_out is poisoned by the harness).
// ---------------------------------------------------------------------------
__global__ void gcn_zero(float* __restrict__ p, int count) {
    int i = blockIdx.x * blockDim.x + threadIdx.x;
    if (i < count) p[i] = 0.0f;
}

// ---------------------------------------------------------------------------
// Kernel 2: edge gather-scale-scatter. 16 lanes per edge (one per feature):
// h[src] row read coalesces to a 64B burst; scatter is atomic f32 add into a
// 6.4MB accumulator that stays resident in the 192MB L2.
// edge_index is int64 per the reference (jnp.int64).
// ---------------------------------------------------------------------------
__global__ void gcn_edge_scatter(const float* __restrict__ h,
                                 const long long* __restrict__ edge_index,
                                 const float* __restrict__ ew,
                                 float* __restrict__ agg) {
    long long idx = (long long)blockIdx.x * blockDim.x + threadIdx.x;
    long long e  = idx >> 4;
    int       f  = (int)(idx & 15);
    if (e >= (long long)N_EDGES) return;
    long long src = edge_index[e];             // row 0
    long long dst = edge_index[N_EDGES + e];   // row 1
    float v = h[src * OUT_F + f] * ew[e];
    atomicAdd(&agg[dst * OUT_F + f], v);
}

// ---------------------------------------------------------------------------
// Kernel 3: out = log_softmax(relu(agg + b)), in place, one thread per node.
// ---------------------------------------------------------------------------
__global__ void gcn_epilogue(float* __restrict__ out, const float* __restrict__ b) {
    int node = blockIdx.x * blockDim.x + threadIdx.x;
    if (node >= N_NODES) return;
    float* row = out + (size_t)node * OUT_F;

    float v[OUT_F];
    float m = -3.402823466e+38f;
#pragma unroll
    for (int f = 0; f < OUT_F; ++f) {
        v[f] = fmaxf(row[f] + b[f], 0.0f);   // bias + relu
        m = fmaxf(m, v[f]);
    }
    float s = 0.0f;
#pragma unroll
    for (int f = 0; f < OUT_F; ++f)
        s += __expf(v[f] - m);
    float lse = m + __logf(s);
#pragma unroll
    for (int f = 0; f < OUT_F; ++f)
        row[f] = v[f] - lse;
}

// ---------------------------------------------------------------------------
extern "C" void kernel_launch(void* const* d_in, const int* in_sizes, int n_in,
                              void* d_out, int out_size, void* d_ws, size_t ws_size,
                              hipStream_t stream) {
    const float*     x          = (const float*)d_in[0];      // [100000,64]
    const long long* edge_index = (const long long*)d_in[1];  // [2,3200000] int64
    const float*     ew         = (const float*)d_in[2];      // [3200000]
    const float*     W          = (const float*)d_in[3];      // [64,16]
    const float*     b          = (const float*)d_in[4];      // [16]
    float* out = (float*)d_out;                               // [100000,16]
    float* h   = (float*)d_ws;                                // 6.4 MB scratch

    // 0) zero the accumulator living in d_out
    gcn_zero<<<(N_NODES * OUT_F + 255) / 256, 256, 0, stream>>>(out, N_NODES * OUT_F);

    // 1) h = x @ W  (WMMA f32; 4 waves/block -> 64 nodes per block)
    gcn_gemm_wmma<<<(NODE_TILES + 3) / 4, 128, 0, stream>>>(x, W, h);

    // 2) scatter-add over edges (16 lanes per edge)
    long long total = (long long)N_EDGES * OUT_F;
    gcn_edge_scatter<<<(unsigned)((total + 255) / 256), 256, 0, stream>>>(h, edge_index, ew, out);

    // 3) bias + relu + log_softmax, in place
    gcn_epilogue<<<(N_NODES + 255) / 256, 256, 0, stream>>>(out, b);
}